// AdaFaceLoss_63110249447794
// MI455X (gfx1250) — compile-verified
//
#include <hip/hip_runtime.h>
#include <math.h>

// ---------------- problem constants (fixed by the reference) ----------------
#define B_SZ   512
#define C_SZ   100000
#define TILE   4096                 // floats per TDM chunk (16 KB)
#define NCH    25                   // ceil(100000 / 4096)
#define LAST   1696                 // 100000 - 24*4096
#define PARTS  4                    // blocks per row (partial sums are additive)
#define TPB    256                  // 8 waves (wave32)

#define H_PARAM 0.333f
#define S_PARAM 64.0f
#define M_PARAM 0.4f
#define EPS_F   1e-6f
#define PI_F    3.14159265358979323846f
#define L2E_F   1.4426950408889634f          // log2(e)
#define K_SCALE (S_PARAM * L2E_F)            // 64*log2(e): exp(64*(x-1)) = 2^((x-1)*K)

// ---------------- CDNA5 Tensor Data Mover (TDM) path ------------------------
#if __has_builtin(__builtin_amdgcn_tensor_load_to_lds) && __has_builtin(__builtin_amdgcn_s_wait_tensorcnt)
#define HAVE_TDM 1
#else
#define HAVE_TDM 0
#endif

typedef __attribute__((ext_vector_type(4))) unsigned int u32x4;
typedef __attribute__((ext_vector_type(8))) int          i32x8;
typedef __attribute__((ext_vector_type(4))) int          i32x4;

#if HAVE_TDM
// Issue a 1-D TDM tile load: nelem f32 from gp -> LDS byte offset lds_byte.
// D# per CDNA5 ISA ch.8: group0 {count=1, lds_addr, global_addr, type=2},
// group1 {data_size=4B, tensor_dim0=nelem, tensor_dim1=1, tile_dim0=nelem,
//         tile_dim1=0 (1-D), tensor_dim0_stride=nelem}; remaining groups zero.
// This toolchain exposes the 6-arg builtin:
//   (uint32x4, int32x8, int32x4, int32x4, int32x8, i32 cpol)
__device__ __forceinline__ void tdm_issue(unsigned lds_byte, const float* gp, unsigned nelem) {
  unsigned long long ga = (unsigned long long)(uintptr_t)gp;
  u32x4 g0 = { 1u,                                             // [1:0]=count=1
               lds_byte,                                       // [63:32]=lds_addr
               (unsigned)ga,                                   // global_addr[31:0]
               (unsigned)((ga >> 32) & 0x1FFFFFFull) | (2u << 30) }; // [56:32] | type=2
  i32x8 g1 = { (int)(2u << 16),                                // data_size = 4 bytes
               (int)((nelem & 0xFFFFu) << 16),                 // tensor_dim0[15:0] @ bits63:48
               (int)(((nelem >> 16) & 0xFFFFu) | (1u << 16)),  // dim0 hi | tensor_dim1=1
               (int)((nelem & 0xFFFFu) << 16),                 // tensor_dim1 hi=0 | tile_dim0
               0,                                              // tile_dim1=0, tile_dim2=0
               (int)nelem,                                     // tensor_dim0_stride[31:0]
               0, 0 };
  i32x4 z4 = { 0, 0, 0, 0 };
  i32x8 z8 = { 0, 0, 0, 0, 0, 0, 0, 0 };
  __builtin_amdgcn_tensor_load_to_lds(g0, g1, z4, z4, z8, 0);
}
#endif

__device__ __forceinline__ float clip_cos(float v) {
  return fminf(fmaxf(v, -1.0f + EPS_F), 1.0f - EPS_F);
}

// ---------------- kernel A: norm statistics -> per-row g_angular -------------
__global__ void kA_stats(const float* __restrict__ norms, float* __restrict__ g_ang) {
  __shared__ float s1[B_SZ];
  __shared__ float s2[B_SZ];
  const int t = threadIdx.x;
  float x = norms[t];
  x = fminf(fmaxf(x, 0.001f), 100.0f);
  s1[t] = x; s2[t] = x * x;
  __syncthreads();
  for (int off = B_SZ / 2; off > 0; off >>= 1) {
    if (t < off) { s1[t] += s1[t + off]; s2[t] += s2[t + off]; }
    __syncthreads();
  }
  const float mean = s1[0] * (1.0f / (float)B_SZ);
  const float var  = (s2[0] - (float)B_SZ * mean * mean) * (1.0f / (float)(B_SZ - 1));
  const float sd   = sqrtf(fmaxf(var, 0.0f));
  float ms = (x - mean) / (sd + EPS_F) * H_PARAM;
  ms = fminf(fmaxf(ms, -1.0f), 1.0f);
  g_ang[t] = -M_PARAM * ms;
}

// ---------------- kernel B: streamed partial sum-of-exp per (row, part) ------
// Fixed-max LSE with M0 = 64 (valid: all scaled inputs < 64 by hard clip).
__global__ void kB_partial(const float* __restrict__ logits, float* __restrict__ part) {
  __shared__ float4 buf4[2][TILE / 4];     // 32 KB double buffer
  __shared__ float  red[TPB];
  const int bid = blockIdx.x;
  const int row = bid / PARTS;
  const int p   = bid % PARTS;
  const int t   = threadIdx.x;
  const float* rowp = logits + (size_t)row * C_SZ;
  const bool w0 = (t >> 5) == 0;           // wave-uniform: only wave 0 drives the TDM
  const unsigned lbase = (unsigned)(uintptr_t)&buf4[0][0];

#if HAVE_TDM
  if (w0) tdm_issue(lbase, rowp + (size_t)p * TILE, TILE);  // chunk p is never the tail
#endif

  float s = 0.0f;
  int slot = 0;
  for (int i = p; i < NCH; i += PARTS) {
    const int len = (i == NCH - 1) ? LAST : TILE;
#if HAVE_TDM
    if (w0) {
      const int nx = i + PARTS;
      if (nx < NCH) {
        const unsigned ln = (nx == NCH - 1) ? LAST : TILE;
        tdm_issue(lbase + (unsigned)(slot ^ 1) * (TILE * 4), rowp + (size_t)nx * TILE, ln);
        __builtin_amdgcn_s_wait_tensorcnt(1);   // current chunk's DMA complete
      } else {
        __builtin_amdgcn_s_wait_tensorcnt(0);
      }
    }
    __syncthreads();
#else
    {
      const float4* src = (const float4*)(rowp + (size_t)i * TILE);
      const int n4c = len >> 2;
      for (int k = t; k < n4c; k += TPB) buf4[slot][k] = src[k];
    }
    __syncthreads();
#endif
    const int n4 = len >> 2;                 // 1024, tail 424 (1696/4 exact)
    for (int k = t; k < n4; k += TPB) {
      const float4 v = buf4[slot][k];
      const float a = clip_cos(v.x), b = clip_cos(v.y);
      const float c = clip_cos(v.z), d = clip_cos(v.w);
      s += exp2f((a - 1.0f) * K_SCALE);
      s += exp2f((b - 1.0f) * K_SCALE);
      s += exp2f((c - 1.0f) * K_SCALE);
      s += exp2f((d - 1.0f) * K_SCALE);
    }
    __syncthreads();                         // buffer reusable before next prefetch
    slot ^= 1;
  }

  red[t] = s;
  __syncthreads();
  for (int off = TPB / 2; off > 0; off >>= 1) {
    if (t < off) red[t] += red[t + off];
    __syncthreads();
  }
  if (t == 0) part[bid] = red[0];
}

// ---------------- kernel C: label-column correction, NLL, batch mean ---------
__global__ void kC_final(const float* __restrict__ logits, const int* __restrict__ labels,
                         const float* __restrict__ g_ang, const float* __restrict__ part,
                         float* __restrict__ out) {
  __shared__ float red[B_SZ];
  const int r = threadIdx.x;
  float s = part[r * PARTS + 0] + part[r * PARTS + 1]
          + part[r * PARTS + 2] + part[r * PARTS + 3];
  const int   lbl  = labels[r];
  const float xl   = logits[(size_t)r * C_SZ + (size_t)lbl];
  const float xt   = clip_cos(xl);
  const float ga   = g_ang[r];            // g_angular = -M*ms
  const float gadd = M_PARAM - ga;        // M + M*ms
  const float th   = acosf(xt);
  const float thm  = fminf(fmaxf(th + ga, EPS_F), PI_F - EPS_F);
  const float tval = (cosf(thm) - gadd) * S_PARAM;
  // swap the label column's plain term for the margin-modified one
  s += exp2f((tval - S_PARAM) * L2E_F) - exp2f((xt - 1.0f) * K_SCALE);
  const float nll = S_PARAM + logf(s) - tval;   // -(tval - (64 + ln s))
  red[r] = nll;
  __syncthreads();
  for (int off = B_SZ / 2; off > 0; off >>= 1) {
    if (r < off) red[r] += red[r + off];
    __syncthreads();
  }
  if (r == 0) out[0] = red[0] * (1.0f / (float)B_SZ);
}

// ---------------- host launcher ---------------------------------------------
extern "C" void kernel_launch(void* const* d_in, const int* in_sizes, int n_in,
                              void* d_out, int out_size, void* d_ws, size_t ws_size,
                              hipStream_t stream) {
  (void)in_sizes; (void)n_in; (void)out_size; (void)ws_size;
  const float* logits = (const float*)d_in[0];
  const float* norms  = (const float*)d_in[1];
  const int*   labels = (const int*)d_in[2];
  float* out = (float*)d_out;
  float* ws  = (float*)d_ws;
  float* g_ang = ws;               // 512 floats
  float* part  = ws + B_SZ;        // 2048 floats

  kA_stats  <<<1,              B_SZ, 0, stream>>>(norms, g_ang);
  kB_partial<<<B_SZ * PARTS,   TPB,  0, stream>>>(logits, part);
  kC_final  <<<1,              B_SZ, 0, stream>>>(logits, labels, g_ang, part, out);
}